// AFPNNet_82910048682518
// MI455X (gfx1250) — compile-verified
//
#include <hip/hip_runtime.h>
#include <math.h>

// ---------------------------------------------------------------------------
// AFPNNet (GAT + GCN encoder, decoder, discriminator) for MI455X / gfx1250.
// Dense GEMMs: V_WMMA_F32_16X16X4_F32, one wave = 16-row strip x full N,
// branch-free inner loop, 32-bit offset induction (GVS addressing).
// Edge aggregation: int32 indices + global_atomic_add_f32 into L2-resident
// [N,64] buffers (25.6MB each, fits the 192MB L2).
// ---------------------------------------------------------------------------

typedef float v2f __attribute__((ext_vector_type(2)));
typedef float v8f __attribute__((ext_vector_type(8)));

#define HIDF 64

// ---- monotonic float<->uint key for atomic segment-max --------------------
__device__ __forceinline__ unsigned fkey_encode(float x) {
    unsigned u = __float_as_uint(x);
    return (u & 0x80000000u) ? ~u : (u | 0x80000000u);
}
__device__ __forceinline__ float fkey_decode(unsigned k) {
    return (k & 0x80000000u) ? __uint_as_float(k ^ 0x80000000u)
                             : __uint_as_float(~k);
}
#define NEG_MAX_KEY 0x00800000u   // fkey_encode(-FLT_MAX)

// ---- fills ----------------------------------------------------------------
__global__ void k_fill_f32(float* __restrict__ p, float v, int n) {
    int i = blockIdx.x * blockDim.x + threadIdx.x;
    if (i < n) p[i] = v;
}
__global__ void k_fill_u32(unsigned* __restrict__ p, unsigned v, int n) {
    int i = blockIdx.x * blockDim.x + threadIdx.x;
    if (i < n) p[i] = v;
}

// ---- WMMA fp32 GEMM -------------------------------------------------------
// C[M, NT*16] = act(A[M,K] @ B[K, NT*16] + bias), C row stride ldc, column
// offset colOff. One wave owns a 16-row M strip and all NT 16-col tiles:
// per K-step: one float2 A load feeds NT independent WMMAs.
// mask != null (only with bias==null, act==0): zero masked output rows,
// exactly equivalent to zeroing the masked input rows of A.
template <int NT>
__global__ void k_gemm_wmma(const float* __restrict__ A,
                            const float* __restrict__ B,
                            float* __restrict__ C,
                            int M, int K, int ldc, int colOff,
                            const int* __restrict__ mask,
                            const float* __restrict__ bias, int act) {
    const int lane  = threadIdx.x & 31;
    const int wave  = threadIdx.x >> 5;
    const int lo    = lane & 15;   // 0..15
    const int hi    = lane >> 4;   // 0/1 : K-half within a 4-step
    const int mTile = blockIdx.x * (blockDim.x >> 5) + wave;
    const int mBase = mTile * 16;
    if (mBase >= M) return;        // wave-uniform exit
    const int row  = mBase + lo;
    const int rowC = (row < M) ? row : (M - 1);   // clamped, loads never guarded

    v8f acc[NT] = {};

    int aoff = rowC * K + 2 * hi;            // A[row, k0 + 2*hi .. +1]
    int boff = (2 * hi) * (NT * 16) + lo;    // B[k0 + 2*hi, tile*16 + lo]
    for (int k0 = 0; k0 < K; k0 += 4) {
        const v2f a = *reinterpret_cast<const v2f*>(A + aoff);
#pragma unroll
        for (int t = 0; t < NT; ++t) {
            v2f b;
            b.x = B[boff + t * 16];
            b.y = B[boff + NT * 16 + t * 16];
            acc[t] = __builtin_amdgcn_wmma_f32_16x16x4_f32(
                /*neg_a=*/false, a, /*neg_b=*/false, b,
                /*c_mod=*/(short)0, acc[t], /*reuse_a=*/false, /*reuse_b=*/false);
        }
        aoff += 4;
        boff += 4 * NT * 16;
    }

    const int cm0 = mBase + hi * 8;          // C/D layout: VGPR r -> M = r + 8*hi
#pragma unroll
    for (int r = 0; r < 8; ++r) {
        const int m = cm0 + r;
        if (m >= M) continue;
        const bool z = (mask != nullptr) && (mask[m] == 1);
        float* __restrict__ crow = C + (size_t)m * ldc + colOff;
#pragma unroll
        for (int t = 0; t < NT; ++t) {
            const int cn = t * 16 + lo;
            float v = z ? 0.f : acc[t][r];
            if (bias) v += bias[cn];
            if (act == 1)      v = fmaxf(v, 0.f);            // relu
            else if (act == 2) v = 1.f / (1.f + expf(-v));   // sigmoid
            crow[cn] = v;
        }
    }
}

// ---- int64 edge_index -> int32 src/dst ------------------------------------
__global__ void k_prep_edges(const long long* __restrict__ e64,
                             int* __restrict__ s32, int* __restrict__ d32,
                             int Etot) {
    int i = blockIdx.x * blockDim.x + threadIdx.x;
    if (i >= Etot) return;
    s32[i] = (int)e64[i];
    d32[i] = (int)e64[i + Etot];
}

// ---- per-node attention dots ----------------------------------------------
__global__ void k_attn_dots(const float* __restrict__ h,
                            const float* __restrict__ av_s,
                            const float* __restrict__ av_d,
                            float* __restrict__ outS,
                            float* __restrict__ outD, int Nn) {
    int n = blockIdx.x * blockDim.x + threadIdx.x;
    if (n >= Nn) return;
    const float* __restrict__ hr = h + (size_t)n * HIDF;
    float s = 0.f, d = 0.f;
#pragma unroll
    for (int f = 0; f < HIDF; ++f) { float v = hr[f]; s += v * av_s[f]; d += v * av_d[f]; }
    outS[n] = s;
    outD[n] = d;
}

// ---- edge logits + segment max (leaky_relu, slope 0.2) --------------------
__global__ void k_edge_logits(const int* __restrict__ s32, const int* __restrict__ d32,
                              const float* __restrict__ as_, const float* __restrict__ ad_,
                              float* __restrict__ e, unsigned* __restrict__ mkey, int Etot) {
    int i = blockIdx.x * blockDim.x + threadIdx.x;
    if (i >= Etot) return;
    float v = as_[s32[i]] + ad_[d32[i]];
    v = (v >= 0.f) ? v : 0.2f * v;
    e[i] = v;
    atomicMax(&mkey[d32[i]], fkey_encode(v));
}

// ---- ex = exp(e - m[dst]) in place; denom[dst] += ex ----------------------
__global__ void k_edge_expsum(const int* __restrict__ d32, float* __restrict__ e,
                              const unsigned* __restrict__ mkey,
                              float* __restrict__ denom, int Etot) {
    int i = blockIdx.x * blockDim.x + threadIdx.x;
    if (i >= Etot) return;
    const int d  = d32[i];
    const float ex = expf(e[i] - fkey_decode(mkey[d]));
    e[i] = ex;
    atomicAdd(&denom[d], ex);
}

// ---- coef = ex / max(denom[dst], 1e-16), in place -------------------------
__global__ void k_coef(const int* __restrict__ d32, float* __restrict__ ex,
                       const float* __restrict__ denom, int Etot) {
    int i = blockIdx.x * blockDim.x + threadIdx.x;
    if (i >= Etot) return;
    ex[i] = ex[i] / fmaxf(denom[d32[i]], 1e-16f);
}

// ---- agg[dst,:] += h[src,:] * cf[edge] ; 16 threads/edge, float4 each -----
__global__ void k_edge_aggregate(const int* __restrict__ s32, const int* __restrict__ d32,
                                 const float* __restrict__ h, const float* __restrict__ cf,
                                 float* __restrict__ agg, int Etot) {
    int t = blockIdx.x * blockDim.x + threadIdx.x;
    int i = t >> 4;
    int q = t & 15;
    if (i >= Etot) return;
    const int s = s32[i], d = d32[i];
    const float c = cf[i];
    const float4 hv = *reinterpret_cast<const float4*>(h + (size_t)s * HIDF + q * 4);
    float* ap = agg + (size_t)d * HIDF + q * 4;
    atomicAdd(ap + 0, hv.x * c);
    atomicAdd(ap + 1, hv.y * c);
    atomicAdd(ap + 2, hv.z * c);
    atomicAdd(ap + 3, hv.w * c);
}

// ---- GCN degree / dinv / edge norm ----------------------------------------
__global__ void k_deg(const int* __restrict__ d32, float* __restrict__ deg, int Etot) {
    int i = blockIdx.x * blockDim.x + threadIdx.x;
    if (i >= Etot) return;
    atomicAdd(&deg[d32[i]], 1.f);
}
__global__ void k_dinv(float* __restrict__ deg, int Nn) {
    int n = blockIdx.x * blockDim.x + threadIdx.x;
    if (n >= Nn) return;
    const float d = deg[n];
    deg[n] = (d > 0.f) ? (1.f / sqrtf(d)) : 0.f;
}
__global__ void k_norm(const int* __restrict__ s32, const int* __restrict__ d32,
                       const float* __restrict__ dinv, float* __restrict__ nrm, int Etot) {
    int i = blockIdx.x * blockDim.x + threadIdx.x;
    if (i >= Etot) return;
    nrm[i] = dinv[s32[i]] * dinv[d32[i]];
}

// ---- out[n, colOff+f] = act(in[n*F+f] + bias[f]);  act: 0=none 1=relu 3=elu
__global__ void k_bias_act_store(const float* __restrict__ in,
                                 const float* __restrict__ bias,
                                 float* __restrict__ out,
                                 int Nn, int F, int ldo, int colOff, int act) {
    int idx = blockIdx.x * blockDim.x + threadIdx.x;
    if (idx >= Nn * F) return;
    const int n = idx / F;
    const int f = idx - n * F;
    float v = in[idx] + (bias ? bias[f] : 0.f);
    if (act == 1)      v = fmaxf(v, 0.f);
    else if (act == 3) v = (v > 0.f) ? v : (expf(v) - 1.f);
    out[(size_t)n * ldo + colOff + f] = v;
}

// ---- discriminator head: disc[n] = d2[n,:32] . W + b ----------------------
__global__ void k_disc(const float* __restrict__ d2, const float* __restrict__ W,
                       const float* __restrict__ b, float* __restrict__ out, int Nn) {
    int n = blockIdx.x * blockDim.x + threadIdx.x;
    if (n >= Nn) return;
    const float* __restrict__ r = d2 + (size_t)n * 32;
    float sum = 0.f;
#pragma unroll
    for (int i = 0; i < 32; ++i) sum += r[i] * W[i];
    out[n] = sum + b[0];
}

// ---------------------------------------------------------------------------
extern "C" void kernel_launch(void* const* d_in, const int* in_sizes, int n_in,
                              void* d_out, int out_size, void* d_ws, size_t ws_size,
                              hipStream_t stream) {
    (void)n_in; (void)out_size; (void)ws_size;

    const float*     feats = (const float*)d_in[0];
    const long long* eidx  = (const long long*)d_in[1];
    const int*       nmask = (const int*)d_in[2];
    const float *W_gat1 = (const float*)d_in[3],  *a_src1 = (const float*)d_in[4],
                *a_dst1 = (const float*)d_in[5],  *b_gat1 = (const float*)d_in[6];
    const float *W_gat2 = (const float*)d_in[7],  *a_src2 = (const float*)d_in[8],
                *a_dst2 = (const float*)d_in[9],  *b_gat2 = (const float*)d_in[10];
    const float *W_gcn1 = (const float*)d_in[11], *b_gcn1 = (const float*)d_in[12];
    const float *W_gcn2 = (const float*)d_in[13], *b_gcn2 = (const float*)d_in[14];
    const float *W_dec  = (const float*)d_in[15], *b_dec  = (const float*)d_in[16];
    const float *W_d1   = (const float*)d_in[17], *b_d1   = (const float*)d_in[18];
    const float *W_d2   = (const float*)d_in[19], *b_d2   = (const float*)d_in[20];
    const float *W_d3   = (const float*)d_in[21], *b_d3   = (const float*)d_in[22];

    const int N      = in_sizes[2];       // 100000
    const int E      = in_sizes[1] / 2;   // 1600000
    const int IN_DIM = in_sizes[0] / N;   // 128

    // ---- workspace carve ----
    float*    hA    = (float*)d_ws;                 // [N,64]
    float*    hB    = hA + (size_t)N * 64;          // [N,64]  agg target
    float*    hC    = hB + (size_t)N * 64;          // [N,64]
    float*    aS    = hC + (size_t)N * 64;          // [N]
    float*    aD    = aS + N;                       // [N]
    unsigned* mkey  = (unsigned*)(aD + N);          // [N]
    float*    denom = (float*)(mkey + N);           // [N]
    float*    degv  = denom + N;                    // [N] -> dinv in place
    float*    exbuf = degv + N;                     // [E] logits -> ex -> coef
    float*    nrm   = exbuf + E;                    // [E] gcn edge norm
    int*      s32   = (int*)(nrm + E);              // [E]
    int*      d32   = s32 + E;                      // [E]

    float* recon = (float*)d_out;                   // [N,128]
    float* enc   = recon + (size_t)N * 128;         // [N,128]
    float* disc  = recon + (size_t)N * 256;         // [N,1]

    const int TB   = 256;
    const int gN   = (N + TB - 1) / TB;
    const int gE   = (E + TB - 1) / TB;
    const int gNF  = (N * 64 + TB - 1) / TB;
    const int gAgg = (E * 16 + TB - 1) / TB;
    const int nMT  = (N + 15) / 16;                 // 16-row M strips
    const int gMM  = (nMT + 7) / 8;                 // 8 waves / block

    k_prep_edges<<<gE, TB, 0, stream>>>(eidx, s32, d32, E);

    // ==================== GAT branch ====================
    k_gemm_wmma<4><<<gMM, TB, 0, stream>>>(feats, W_gat1, hA, N, IN_DIM,
                                           64, 0, nmask, nullptr, 0);
    k_attn_dots<<<gN, TB, 0, stream>>>(hA, a_src1, a_dst1, aS, aD, N);
    k_fill_u32<<<gN, TB, 0, stream>>>(mkey, NEG_MAX_KEY, N);
    k_fill_f32<<<gN, TB, 0, stream>>>(denom, 0.f, N);
    k_edge_logits<<<gE, TB, 0, stream>>>(s32, d32, aS, aD, exbuf, mkey, E);
    k_edge_expsum<<<gE, TB, 0, stream>>>(d32, exbuf, mkey, denom, E);
    k_coef<<<gE, TB, 0, stream>>>(d32, exbuf, denom, E);
    k_fill_f32<<<gNF, TB, 0, stream>>>(hB, 0.f, N * 64);
    k_edge_aggregate<<<gAgg, TB, 0, stream>>>(s32, d32, hA, exbuf, hB, E);
    k_bias_act_store<<<gNF, TB, 0, stream>>>(hB, b_gat1, hA, N, 64, 64, 0, 3); // elu

    k_gemm_wmma<4><<<gMM, TB, 0, stream>>>(hA, W_gat2, hC, N, 64,
                                           64, 0, nullptr, nullptr, 0);
    k_attn_dots<<<gN, TB, 0, stream>>>(hC, a_src2, a_dst2, aS, aD, N);
    k_fill_u32<<<gN, TB, 0, stream>>>(mkey, NEG_MAX_KEY, N);
    k_fill_f32<<<gN, TB, 0, stream>>>(denom, 0.f, N);
    k_edge_logits<<<gE, TB, 0, stream>>>(s32, d32, aS, aD, exbuf, mkey, E);
    k_edge_expsum<<<gE, TB, 0, stream>>>(d32, exbuf, mkey, denom, E);
    k_coef<<<gE, TB, 0, stream>>>(d32, exbuf, denom, E);
    k_fill_f32<<<gNF, TB, 0, stream>>>(hB, 0.f, N * 64);
    k_edge_aggregate<<<gAgg, TB, 0, stream>>>(s32, d32, hC, exbuf, hB, E);
    k_bias_act_store<<<gNF, TB, 0, stream>>>(hB, b_gat2, enc, N, 64, 128, 0, 0);

    // ==================== GCN branch ====================
    k_fill_f32<<<gN, TB, 0, stream>>>(degv, 0.f, N);
    k_deg<<<gE, TB, 0, stream>>>(d32, degv, E);
    k_dinv<<<gN, TB, 0, stream>>>(degv, N);
    k_norm<<<gE, TB, 0, stream>>>(s32, d32, degv, nrm, E);

    k_gemm_wmma<4><<<gMM, TB, 0, stream>>>(feats, W_gcn1, hA, N, IN_DIM,
                                           64, 0, nmask, nullptr, 0);
    k_fill_f32<<<gNF, TB, 0, stream>>>(hB, 0.f, N * 64);
    k_edge_aggregate<<<gAgg, TB, 0, stream>>>(s32, d32, hA, nrm, hB, E);
    k_bias_act_store<<<gNF, TB, 0, stream>>>(hB, b_gcn1, hA, N, 64, 64, 0, 1); // relu

    k_gemm_wmma<4><<<gMM, TB, 0, stream>>>(hA, W_gcn2, hC, N, 64,
                                           64, 0, nullptr, nullptr, 0);
    k_fill_f32<<<gNF, TB, 0, stream>>>(hB, 0.f, N * 64);
    k_edge_aggregate<<<gAgg, TB, 0, stream>>>(s32, d32, hC, nrm, hB, E);
    k_bias_act_store<<<gNF, TB, 0, stream>>>(hB, b_gcn2, enc, N, 64, 128, 64, 0);

    // ==================== heads ====================
    k_gemm_wmma<8><<<gMM, TB, 0, stream>>>(enc, W_dec, recon, N, 128,
                                           128, 0, nullptr, b_dec, 1);   // relu
    k_gemm_wmma<4><<<gMM, TB, 0, stream>>>(enc, W_d1, hA, N, 128,
                                           64, 0, nullptr, b_d1, 1);     // relu
    k_gemm_wmma<2><<<gMM, TB, 0, stream>>>(hA, W_d2, hB, N, 64,
                                           32, 0, nullptr, b_d2, 2);     // sigmoid
    k_disc<<<gN, TB, 0, stream>>>(hB, W_d3, b_d3, disc, N);
}